// MultiHeadAttention_29180007809486
// MI455X (gfx1250) — compile-verified
//
#include <hip/hip_runtime.h>
#include <cmath>

// Problem sizes (fixed by the reference)
#define P_DIM 4096
#define B_DIM 32
#define FEATN 512
#define ATTN  512
#define NHEAD 8
#define HD    64

typedef __attribute__((ext_vector_type(2))) float v2f;
typedef __attribute__((ext_vector_type(8))) float v8f;
typedef __attribute__((ext_vector_type(4))) int v4i;

// CDNA5 fp32 WMMA: D(16x16,f32) = A(16x4,f32) * B(4x16,f32) + C
__device__ __forceinline__ v8f wmma_f32x4(v2f a, v2f b, v8f c) {
  return __builtin_amdgcn_wmma_f32_16x16x4_f32(false, a, false, b, (short)0, c,
                                               false, false);
}

// ---------------------------------------------------------------------------
// Async global->LDS staging (CDNA5 GLOBAL_LOAD_ASYNC_TO_LDS_B128, ASYNCcnt).
// Confirmed present on this toolchain; signature (from round-2 diagnostic):
//   (v4i addrspace(1)*, v4i addrspace(3)*, imm offset, imm cpol)
// ---------------------------------------------------------------------------
#if __has_builtin(__builtin_amdgcn_global_load_async_to_lds_b128) && \
    __has_builtin(__builtin_amdgcn_s_wait_asynccnt)
#define HAS_ASYNC_LDS 1
typedef __attribute__((address_space(1))) v4i g_v4i;
typedef __attribute__((address_space(3))) v4i l_v4i;
#else
#define HAS_ASYNC_LDS 0
#endif

__device__ __forceinline__ void copy16_g2lds(const float4* g, float4* l) {
#if HAS_ASYNC_LDS
  __builtin_amdgcn_global_load_async_to_lds_b128((g_v4i*)g, (l_v4i*)l, 0, 0);
#else
  *l = *g;
#endif
}

__device__ __forceinline__ void wait_async_fills() {
#if HAS_ASYNC_LDS
  __builtin_amdgcn_s_wait_asynccnt(0);
#endif
}

// ---------------------------------------------------------------------------
// Kernel A: hidterm[b][h*64+d] = (last_hidden[b] @ Ua + Ua_b)[head] @ Ua2 + Ua2_b
// Tiny (32 rows) -> plain VALU.
// ---------------------------------------------------------------------------
__global__ __launch_bounds__(512) void prep_hidterm_kernel(
    const float* __restrict__ last_hidden,  // [B,512]
    const float* __restrict__ Ua,           // [512,512] (in,out)
    const float* __restrict__ Uab,          // [512]
    const float* __restrict__ Ua2,          // [64,64]
    const float* __restrict__ Ua2b,         // [64]
    float* __restrict__ hidterm)            // [B,512]
{
  __shared__ float lh[FEATN];
  __shared__ float hid[ATTN];
  const int b = blockIdx.x;
  const int a = threadIdx.x;
  lh[a] = last_hidden[b * FEATN + a];
  __syncthreads();
  float s = Uab[a];
  for (int h2 = 0; h2 < FEATN; ++h2) s += lh[h2] * Ua[h2 * ATTN + a];
  hid[a] = s;
  __syncthreads();
  const int head = a >> 6, d = a & 63;
  float s2 = Ua2b[d];
  for (int j = 0; j < HD; ++j) s2 += hid[head * HD + j] * Ua2[j * HD + d];
  hidterm[b * ATTN + a] = s2;
}

// ---------------------------------------------------------------------------
// Kernel B (main, WMMA): per WG handle 64 rows (r = p*B+b) of:
//   enc = X @ Wa + Wa_b                              (f32 WMMA, K=512)
//   a   = tanh(enc_h @ Wa2 + Wa2_b + hidterm[b,h])   (f32 WMMA, K=64, per head)
//   score[b,h,p] = a . va + va_b
// X block (async-filled) and enc block both live in LDS (2 x 128 KB of the
// 320 KB WGP LDS).
// ---------------------------------------------------------------------------
__global__ __launch_bounds__(256) void fused_proj_score_kernel(
    const float* __restrict__ X,        // [P*B, 512]
    const float* __restrict__ Wa,       // [512,512] (K,N)
    const float* __restrict__ Wab,      // [512]
    const float* __restrict__ Wa2,      // [64,64]   (K,N)
    const float* __restrict__ Wa2b,     // [64]
    const float* __restrict__ va,       // [512] (64 per head)
    const float* __restrict__ vab,      // [1]
    const float* __restrict__ hidterm,  // [B,512]
    float* __restrict__ enc_g,          // [P*B,512]
    float* __restrict__ scores)         // [B*NHEAD, P]
{
  extern __shared__ float sm[];
  float* Xlds = sm;                 // 64 x 512
  float* Elds = sm + 64 * FEATN;    // 64 x 512

  const int rowBase = blockIdx.x * 64;
  const int tid  = threadIdx.x;
  const int lane = tid & 31;
  const int wave = tid >> 5;
  const int m    = lane & 15;   // M for A/C, N for B/C
  const int hi   = lane >> 4;   // lane half
  const int koff = hi * 2;      // K sub-offset for A/B fragments

  // Stage 64x512 X rows into LDS: async direct global->LDS (no VGPR bounce)
  {
    const float4* src = (const float4*)(X + (size_t)rowBase * FEATN);
    float4* dst = (float4*)Xlds;
#pragma unroll 4
    for (int i = tid; i < 64 * FEATN / 4; i += 256) copy16_g2lds(src + i, dst + i);
    wait_async_fills();
  }
  __syncthreads();

  const int wr = wave >> 1;  // row group 0..3  -> rows wr*16..+16
  const int wc = wave & 1;   // col half 0..1

  // ---- big GEMM: enc = X @ Wa + Wab, 4 passes of 128 cols ----
  for (int nc = 0; nc < 4; ++nc) {
    const int colBase = nc * 128 + wc * 64;
    v8f acc[4] = {};
    for (int k = 0; k < FEATN; k += 4) {
      v2f a;
      const float* xr = &Xlds[(wr * 16 + m) * FEATN + k + koff];
      a.x = xr[0];
      a.y = xr[1];
#pragma unroll
      for (int t = 0; t < 4; ++t) {
        const int col = colBase + t * 16 + m;
        v2f bf;
        bf.x = Wa[(size_t)(k + koff) * ATTN + col];
        bf.y = Wa[(size_t)(k + koff + 1) * ATTN + col];
        acc[t] = wmma_f32x4(a, bf, acc[t]);
      }
    }
#pragma unroll
    for (int t = 0; t < 4; ++t) {
      const int col  = colBase + t * 16 + m;
      const float bi = Wab[col];
#pragma unroll
      for (int i = 0; i < 8; ++i) {
        const int rl  = wr * 16 + hi * 8 + i;
        const float v = acc[t][i] + bi;
        enc_g[(size_t)(rowBase + rl) * ATTN + col] = v;
        Elds[rl * ATTN + col] = v;
      }
    }
  }
  __syncthreads();

  // ---- inner head attention on the LDS-resident enc block ----
  const int wr2 = wave >> 1;       // same 16-row group
  const int wh  = wave & 1;        // heads wh*4 .. wh*4+3
  const float va_bias = vab[0];

  for (int hh = 0; hh < 4; ++hh) {
    const int h = wh * 4 + hh;
    float sc[8] = {0, 0, 0, 0, 0, 0, 0, 0};
#pragma unroll
    for (int t = 0; t < 4; ++t) {
      const int n = t * 16 + m;            // col within head (0..63)
      v8f acc2;
      const float wb = Wa2b[n];
#pragma unroll
      for (int i = 0; i < 8; ++i) {        // C init = Wa2_b + hidterm
        const int rl = wr2 * 16 + hi * 8 + i;
        const int b  = (rowBase + rl) & (B_DIM - 1);  // r = p*B+b, B=32
        acc2[i] = wb + hidterm[b * ATTN + h * HD + n];
      }
#pragma unroll
      for (int k = 0; k < HD; k += 4) {
        v2f a;
        const float* er = &Elds[(wr2 * 16 + m) * ATTN + h * HD + k + koff];
        a.x = er[0];
        a.y = er[1];
        v2f bf;
        bf.x = Wa2[(k + koff) * HD + n];
        bf.y = Wa2[(k + koff + 1) * HD + n];
        acc2 = wmma_f32x4(a, bf, acc2);
      }
      const float vw = va[h * HD + n];
#pragma unroll
      for (int i = 0; i < 8; ++i) sc[i] += tanhf(acc2[i]) * vw;
    }
    // reduce over 16 lanes of each half (xor 1..8 stays within a half; wave32)
#pragma unroll
    for (int off = 8; off >= 1; off >>= 1) {
#pragma unroll
      for (int i = 0; i < 8; ++i) sc[i] += __shfl_xor(sc[i], off, 32);
    }
    if (m == 0) {  // lane 0 -> rows M=i, lane 16 -> rows M=i+8
#pragma unroll
      for (int i = 0; i < 8; ++i) {
        const int rl = wr2 * 16 + hi * 8 + i;
        const int r  = rowBase + rl;
        const int b  = r & (B_DIM - 1);
        const int p  = r >> 5;
        scores[(size_t)((b * NHEAD + h) << 12) + p] = sc[i] + va_bias;
      }
    }
  }
}

// ---------------------------------------------------------------------------
// Kernel C: softmax over P per (b,h) column.  scores/weights layout [b*8+h][P].
// ---------------------------------------------------------------------------
__global__ __launch_bounds__(256) void softmax_kernel(
    const float* __restrict__ scores, float* __restrict__ weights)
{
  __shared__ float red[256];
  const int col = blockIdx.x;
  const float* s = scores + ((size_t)col << 12);
  float v[16];
  float mx = -3.4e38f;
#pragma unroll
  for (int j = 0; j < 16; ++j) {
    v[j] = s[threadIdx.x + j * 256];
    mx = fmaxf(mx, v[j]);
  }
  red[threadIdx.x] = mx;
  __syncthreads();
  for (int off = 128; off >= 1; off >>= 1) {
    if (threadIdx.x < off)
      red[threadIdx.x] = fmaxf(red[threadIdx.x], red[threadIdx.x + off]);
    __syncthreads();
  }
  mx = red[0];
  __syncthreads();
  float sum = 0.f;
#pragma unroll
  for (int j = 0; j < 16; ++j) { v[j] = __expf(v[j] - mx); sum += v[j]; }
  red[threadIdx.x] = sum;
  __syncthreads();
  for (int off = 128; off >= 1; off >>= 1) {
    if (threadIdx.x < off) red[threadIdx.x] += red[threadIdx.x + off];
    __syncthreads();
  }
  const float inv = 1.0f / red[0];
  float* w = weights + ((size_t)col << 12);
#pragma unroll
  for (int j = 0; j < 16; ++j) w[threadIdx.x + j * 256] = v[j] * inv;
}

// ---------------------------------------------------------------------------
// Kernel D: weight[p,b] = mean over heads of weights[b,h,p]
// ---------------------------------------------------------------------------
__global__ __launch_bounds__(256) void weight_mean_kernel(
    const float* __restrict__ weights, float* __restrict__ w_out)
{
  const int idx = blockIdx.x * 256 + threadIdx.x;  // idx = p*B + b
  const int p = idx >> 5, b = idx & 31;
  float s = 0.f;
#pragma unroll
  for (int h = 0; h < NHEAD; ++h) s += weights[(size_t)((b * NHEAD + h) << 12) + p];
  w_out[idx] = s * 0.125f;
}

// ---------------------------------------------------------------------------
// Kernel E: context[b, h*64+d] = sum_p weights[b,h,p] * enc[p*B+b, h*64+d]
// One block per (b,h): threads = (pg 0..3) x (d 0..63); lanes share p -> coalesced.
// ---------------------------------------------------------------------------
__global__ __launch_bounds__(256) void context_kernel(
    const float* __restrict__ weights, const float* __restrict__ enc,
    float* __restrict__ ctx_out)
{
  __shared__ float red[256];
  const int bh = blockIdx.x;
  const int b = bh >> 3, h = bh & 7;
  const int d  = threadIdx.x & 63;
  const int pg = threadIdx.x >> 6;
  const float* w = weights + ((size_t)(b * NHEAD + h) << 12);
  float s = 0.f;
  const int p0 = pg * (P_DIM / 4);
  for (int p = p0; p < p0 + P_DIM / 4; ++p)
    s += w[p] * enc[((size_t)(p * B_DIM + b) << 9) + h * HD + d];
  red[threadIdx.x] = s;
  __syncthreads();
  if (threadIdx.x < 64) {
    const float t = red[threadIdx.x] + red[threadIdx.x + 64] +
                    red[threadIdx.x + 128] + red[threadIdx.x + 192];
    ctx_out[b * ATTN + h * HD + d] = t;
  }
}

// ---------------------------------------------------------------------------
extern "C" void kernel_launch(void* const* d_in, const int* in_sizes, int n_in,
                              void* d_out, int out_size, void* d_ws, size_t ws_size,
                              hipStream_t stream) {
  (void)in_sizes; (void)n_in; (void)out_size; (void)ws_size;
  const float* last_hidden = (const float*)d_in[0];
  const float* X    = (const float*)d_in[1];
  const float* Wa   = (const float*)d_in[2];
  const float* Wab  = (const float*)d_in[3];
  const float* Ua   = (const float*)d_in[4];
  const float* Uab  = (const float*)d_in[5];
  const float* Wa2  = (const float*)d_in[6];
  const float* Wa2b = (const float*)d_in[7];
  const float* Ua2  = (const float*)d_in[8];
  const float* Ua2b = (const float*)d_in[9];
  const float* va   = (const float*)d_in[10];
  const float* vab  = (const float*)d_in[11];
  // d_in[12] = nhead (hardcoded 8)

  float* ws      = (float*)d_ws;
  float* enc     = ws;                                        // 64M floats (256 MiB)
  float* scores  = enc + (size_t)P_DIM * B_DIM * ATTN;        // 1M floats
  float* weights = scores + (size_t)B_DIM * NHEAD * P_DIM;    // 1M floats
  float* hidterm = weights + (size_t)B_DIM * NHEAD * P_DIM;   // 16K floats

  float* ctx_out = (float*)d_out;            // [1,B,512] = 16384
  float* w_out   = ctx_out + B_DIM * ATTN;   // [P,B,1]   = 131072

  prep_hidterm_kernel<<<dim3(B_DIM), dim3(512), 0, stream>>>(
      last_hidden, Ua, Uab, Ua2, Ua2b, hidterm);

  const size_t lds_bytes = 2u * 64u * FEATN * sizeof(float);  // 256 KB of 320 KB
  fused_proj_score_kernel<<<dim3(P_DIM * B_DIM / 64), dim3(256), lds_bytes, stream>>>(
      X, Wa, Wab, Wa2, Wa2b, va, vab, hidterm, enc, scores);

  softmax_kernel<<<dim3(B_DIM * NHEAD), dim3(256), 0, stream>>>(scores, weights);

  weight_mean_kernel<<<dim3(P_DIM * B_DIM / 256), dim3(256), 0, stream>>>(weights, w_out);

  context_kernel<<<dim3(B_DIM * NHEAD), dim3(256), 0, stream>>>(weights, enc, ctx_out);
}